// SparseMultiRangeAttention2D_22935125360702
// MI455X (gfx1250) — compile-verified
//
#include <hip/hip_runtime.h>
#include <math.h>
#include <stdint.h>

typedef __attribute__((ext_vector_type(2))) float v2f;
typedef __attribute__((ext_vector_type(8))) float v8f;

#define HW    96
#define NPIX  (96 * 96)        // 9216
#define C_IN  192
#define C_QKV 576
#define NB    4

#define USE_ASYNC_LDS 1        // gfx1250 GLOBAL_LOAD_ASYNC_TO_LDS staging path

// Stage a [rows x 192] weight strip (row n0+n) into LDS with stride-193 padding.
// Async variant: each lane copies one dword global->LDS, tracked by ASYNCcnt.
__device__ __forceinline__ void stage_weights(float* Bs, const float* __restrict__ w,
                                              int n0, int rows) {
#if USE_ASYNC_LDS
  for (int idx = threadIdx.x; idx < rows * C_IN; idx += 256) {
    const int n = idx / C_IN, k = idx % C_IN;
    const unsigned lds_off = (unsigned)(size_t)&Bs[n * 193 + k];
    const float* g = &w[(size_t)(n0 + n) * C_IN + k];
    asm volatile("global_load_async_to_lds_b32 %0, %1, off"
                 :: "v"(lds_off), "v"(g) : "memory");
  }
  asm volatile("s_wait_asynccnt 0x0" ::: "memory");
#else
  for (int idx = threadIdx.x; idx < rows * C_IN; idx += 256) {
    const int n = idx / C_IN, k = idx % C_IN;
    Bs[n * 193 + k] = w[(size_t)(n0 + n) * C_IN + k];
  }
#endif
  __syncthreads();
}

// ---------------------------------------------------------------------------
// Kernel 1: qkv[(b*NPIX+pix)][o] = sum_c x[b][c][pix] * w_qkv[o][c]
// x: [B, 192, 9216] (c-major), w_qkv: [576, 192] row-major,
// qkv out: [B*9216, 576] pixel-major (per-pixel q/k/v head vectors = one
// aligned 128B line each, which is what the attention stage wants).
// Per-wave tile: 16(M=pixels) x 32(N=outputs): one A frag feeds two WMMAs.
// A frag (16x4 f32): lane<16 -> M=lane, K={k0,k0+1}; lane>=16 -> K={k0+2,k0+3}.
// B frag (4x16 f32): VGPR0: K=k0 (lanes0-15) / K=k0+2 (lanes16-31); VGPR1: +1.
// ---------------------------------------------------------------------------
__global__ __launch_bounds__(256) void qkv_gemm(const float* __restrict__ x,
                                                const float* __restrict__ w,
                                                float* __restrict__ qkv) {
  __shared__ float Bs[32 * 193];           // stride 193: conflict-free frag reads
  const int n0 = blockIdx.y * 32;
  stage_weights(Bs, w, n0, 32);

  const int wave = threadIdx.x >> 5;
  const int lane = threadIdx.x & 31;
  const int m0   = (blockIdx.x * 8 + wave) * 16;   // global row = b*NPIX + pix0
  const int b    = m0 / NPIX;                      // NPIX % 16 == 0: no straddle
  const int pix  = (m0 % NPIX) + (lane & 15);
  const int kh   = lane >> 4;
  const int nl   = lane & 15;
  const float* xb = x + (size_t)b * C_IN * NPIX;

  v8f acc0 = {}, acc1 = {};
  for (int k0 = 0; k0 < C_IN; k0 += 4) {
    const int ka = k0 + 2 * kh;
    v2f a, b0, b1;
    a.x  = xb[(size_t)ka * NPIX + pix];
    a.y  = xb[(size_t)(ka + 1) * NPIX + pix];
    b0.x = Bs[nl * 193 + ka];
    b0.y = Bs[nl * 193 + ka + 1];
    b1.x = Bs[(nl + 16) * 193 + ka];
    b1.y = Bs[(nl + 16) * 193 + ka + 1];
    acc0 = __builtin_amdgcn_wmma_f32_16x16x4_f32(false, a, false, b0,
                                                 (short)0, acc0, false, false);
    acc1 = __builtin_amdgcn_wmma_f32_16x16x4_f32(false, a, false, b1,
                                                 (short)0, acc1, false, false);
  }

  // D layout: VGPR r -> (M = m0 + r + 8*kh, N = n0 + nl); N contiguous per half.
  float* dst = qkv + (size_t)m0 * C_QKV + n0;
#pragma unroll
  for (int r = 0; r < 8; ++r) {
    dst[(size_t)(r + 8 * kh) * C_QKV + nl]      = acc0[r];
    dst[(size_t)(r + 8 * kh) * C_QKV + 16 + nl] = acc1[r];
  }
}

// ---------------------------------------------------------------------------
// Kernel 2: per-head dilated neighborhood attention, online softmax.
// qkv: [B*NPIX, 576]  (q at +head*32, k at +192+head*32, v at +384+head*32)
// Reference semantics: zero-padded patches -> OOB neighbor score = bias[p],
// OOB V contributes zero, but both stay inside the softmax.
// ---------------------------------------------------------------------------
template <int K, int DIL>
__global__ __launch_bounds__(256) void natten_head(const float* __restrict__ qkv,
                                                   const float* __restrict__ bias,
                                                   float* __restrict__ outbuf,
                                                   int head) {
  const int t = blockIdx.x * 256 + threadIdx.x;     // (b, pix)
  if (t >= NB * NPIX) return;
  const int b = t / NPIX, pix = t % NPIX;
  const int py = pix / HW, px = pix % HW;
  const float scale = 0.17677669529663689f;          // 1/sqrt(32)

  const float* qp = qkv + (size_t)t * C_QKV + head * 32;
  float q[32];
#pragma unroll
  for (int d = 0; d < 32; d += 4) {
    float4 v = *(const float4*)(qp + d);
    q[d] = v.x; q[d + 1] = v.y; q[d + 2] = v.z; q[d + 3] = v.w;
  }

  float m = -INFINITY, ssum = 0.f;
  float acc[32];
#pragma unroll
  for (int d = 0; d < 32; ++d) acc[d] = 0.f;

  for (int i = 0; i < K; ++i) {
    for (int j = 0; j < K; ++j) {
      const int p  = i * K + j;
      const int ny = py + (i - K / 2) * DIL;
      const int nx = px + (j - K / 2) * DIL;
      const bool inb = ((unsigned)ny < HW) && ((unsigned)nx < HW);
      const float* nb = qkv + (size_t)(b * NPIX + ny * HW + nx) * C_QKV
                            + C_IN + head * 32;     // K vector (aligned 128B)
      float s;
      if (inb) {
        float dot = 0.f;
#pragma unroll
        for (int d = 0; d < 32; d += 4) {
          float4 kv = *(const float4*)(nb + d);
          dot += q[d] * kv.x + q[d + 1] * kv.y + q[d + 2] * kv.z + q[d + 3] * kv.w;
        }
        s = dot * scale + bias[p];
      } else {
        s = bias[p];                                 // zero-padded K => dot = 0
      }
      const float mn   = fmaxf(m, s);
      const float cold = __expf(m - mn);             // exp(-inf)=0 on first iter
      const float wgt  = __expf(s - mn);
      ssum = ssum * cold + wgt;
#pragma unroll
      for (int d = 0; d < 32; ++d) acc[d] *= cold;
      if (inb) {
        const float* vv = nb + C_IN;                 // V vector
#pragma unroll
        for (int d = 0; d < 32; d += 4) {
          float4 v4 = *(const float4*)(vv + d);
          acc[d]     += wgt * v4.x;  acc[d + 1] += wgt * v4.y;
          acc[d + 2] += wgt * v4.z;  acc[d + 3] += wgt * v4.w;
        }
      }
      m = mn;
    }
  }

  const float inv = 1.0f / ssum;
  float* op = outbuf + (size_t)t * C_IN + head * 32;
#pragma unroll
  for (int d = 0; d < 32; d += 4) {
    float4 r;
    r.x = acc[d] * inv;     r.y = acc[d + 1] * inv;
    r.z = acc[d + 2] * inv; r.w = acc[d + 3] * inv;
    *(float4*)(op + d) = r;
  }
}

// ---------------------------------------------------------------------------
// Kernel 3: out[b][o][pix] = sum_c attn[(b*NPIX+pix)][c] * w_out[o][c]
// A: [B*9216, 192] row-major, B: [192,192] row-major staged in LDS,
// per-wave 16M x 32N tile, D transposed through per-wave LDS tile for
// coalesced NCHW stores.
// ---------------------------------------------------------------------------
__global__ __launch_bounds__(256) void out_gemm(const float* __restrict__ a_,
                                                const float* __restrict__ w,
                                                float* __restrict__ out) {
  __shared__ float Bs[32 * 193];
  __shared__ float T[8][16 * 33];                    // per-wave transpose tile
  const int n0 = blockIdx.y * 32;
  stage_weights(Bs, w, n0, 32);

  const int wave = threadIdx.x >> 5;
  const int lane = threadIdx.x & 31;
  const int m0   = (blockIdx.x * 8 + wave) * 16;
  const int b    = m0 / NPIX;
  const int pix0 = m0 % NPIX;
  const int kh   = lane >> 4;
  const int nl   = lane & 15;
  const float* arow = a_ + (size_t)(m0 + nl) * C_IN;  // lane's A row

  v8f acc0 = {}, acc1 = {};
  for (int k0 = 0; k0 < C_IN; k0 += 4) {
    const int ka = k0 + 2 * kh;
    v2f a = *(const v2f*)(arow + ka);                 // 8B aligned (ka even)
    v2f b0, b1;
    b0.x = Bs[nl * 193 + ka];
    b0.y = Bs[nl * 193 + ka + 1];
    b1.x = Bs[(nl + 16) * 193 + ka];
    b1.y = Bs[(nl + 16) * 193 + ka + 1];
    acc0 = __builtin_amdgcn_wmma_f32_16x16x4_f32(false, a, false, b0,
                                                 (short)0, acc0, false, false);
    acc1 = __builtin_amdgcn_wmma_f32_16x16x4_f32(false, a, false, b1,
                                                 (short)0, acc1, false, false);
  }

  // D -> LDS (row = pixel-in-tile, col = channel-in-strip), stride 33.
#pragma unroll
  for (int r = 0; r < 8; ++r) {
    T[wave][(r + 8 * kh) * 33 + nl]      = acc0[r];
    T[wave][(r + 8 * kh) * 33 + 16 + nl] = acc1[r];
  }
  // same-wave LDS RAW: compiler inserts s_wait_dscnt

  // Coalesced NCHW store: half-waves cover two channel rows, 16 pixels each.
#pragma unroll
  for (int it = 0; it < 16; ++it) {
    const int rr = 2 * it + kh;                       // channel within strip
    out[((size_t)b * C_IN + n0 + rr) * NPIX + pix0 + nl] = T[wave][nl * 33 + rr];
  }
}

// ---------------------------------------------------------------------------
extern "C" void kernel_launch(void* const* d_in, const int* in_sizes, int n_in,
                              void* d_out, int out_size, void* d_ws, size_t ws_size,
                              hipStream_t stream) {
  const float* x      = (const float*)d_in[0];
  const float* w_qkv  = (const float*)d_in[1];
  const float* w_out  = (const float*)d_in[2];
  const float* bias0  = (const float*)d_in[3];
  const float* bias1  = (const float*)d_in[4];
  const float* bias2  = (const float*)d_in[5];
  const float* bias3  = (const float*)d_in[6];
  const float* bias4  = (const float*)d_in[7];
  const float* bias5  = (const float*)d_in[8];

  float* ws1 = (float*)d_ws;                                  // [B*9216, 576]
  float* ws2 = ws1 + (size_t)NB * NPIX * C_QKV;               // [B*9216, 192]
  float* out = (float*)d_out;                                 // [B, 192, 9216]

  const int mblocks = (NB * NPIX) / (8 * 16);                 // 288

  qkv_gemm<<<dim3(mblocks, C_QKV / 32), 256, 0, stream>>>(x, w_qkv, ws1);

  const int ablocks = (NB * NPIX + 255) / 256;                // 144
  natten_head<3, 1><<<ablocks, 256, 0, stream>>>(ws1, bias0, ws2, 0);
  natten_head<5, 2><<<ablocks, 256, 0, stream>>>(ws1, bias1, ws2, 1);
  natten_head<7, 1><<<ablocks, 256, 0, stream>>>(ws1, bias2, ws2, 2);
  natten_head<7, 3><<<ablocks, 256, 0, stream>>>(ws1, bias3, ws2, 3);
  natten_head<9, 1><<<ablocks, 256, 0, stream>>>(ws1, bias4, ws2, 4);
  natten_head<9, 2><<<ablocks, 256, 0, stream>>>(ws1, bias5, ws2, 5);

  out_gemm<<<dim3(mblocks, C_IN / 32), 256, 0, stream>>>(ws2, w_out, out);
}